// GDELayer_old_39367670235152
// MI455X (gfx1250) — compile-verified
//
#include <hip/hip_runtime.h>

#define D 32  // D_IN == D_OUT == 32

typedef __attribute__((ext_vector_type(2))) float v2f;
typedef __attribute__((ext_vector_type(8))) float v8f;

// ---------------------------------------------------------------------------
// Kernel 1: hw[n,:] = (h[n,:] @ W) * norm[n]   via V_WMMA_F32_16X16X4_F32.
// One wave32 computes a 16x32 tile of hw: 2 N-tiles x 8 K-steps (K=4 each).
// fp32 WMMA keeps exact reference precision.
// ---------------------------------------------------------------------------
__global__ __launch_bounds__(256) void gde_gemm_wmma(
    const float* __restrict__ h,       // [N, 32]
    const float* __restrict__ weight,  // [32, 32] row-major
    const float* __restrict__ norm,    // [N]
    float* __restrict__ hw,            // [N, 32]
    int N)
{
  const int lane    = threadIdx.x & 31;
  const int lane16  = lane & 15;
  const int halfSel = lane >> 4;                       // 0: lanes 0-15, 1: 16-31
  const int tile    = blockIdx.x * (blockDim.x >> 5) + (threadIdx.x >> 5);
  const int numTiles = (N + 15) >> 4;
  if (tile >= numTiles) return;                        // wave-uniform: EXEC stays all-1
  const int row0 = tile << 4;

  const int  rowA     = row0 + lane16;
  const bool rowValid = rowA < N;
  const float* hrow   = h + (size_t)(rowValid ? rowA : (N - 1)) * D;

  // A fragments for all 8 K-steps. ISA 16x4 f32 layout:
  //   lanes 0-15: M=lane, v0=K0 v1=K1 ; lanes 16-31: M=lane-16, v0=K2 v1=K3
  v2f a[8];
#pragma unroll
  for (int k = 0; k < 8; ++k) {
    const float x0 = hrow[4 * k + 2 * halfSel + 0];
    const float x1 = hrow[4 * k + 2 * halfSel + 1];
    a[k].x = rowValid ? x0 : 0.0f;
    a[k].y = rowValid ? x1 : 0.0f;
  }

  v8f acc0 = {};  // output cols 0..15
  v8f acc1 = {};  // output cols 16..31
#pragma unroll
  for (int k = 0; k < 8; ++k) {
    const int kr = 4 * k + 2 * halfSel;
    v2f b0, b1;  // B (4x16): lane holds W[kr..kr+1][col], col = lane16 (+16 for tile 1)
    b0.x = weight[(size_t)kr * D + lane16];
    b0.y = weight[(size_t)(kr + 1) * D + lane16];
    b1.x = weight[(size_t)kr * D + 16 + lane16];
    b1.y = weight[(size_t)(kr + 1) * D + 16 + lane16];
    acc0 = __builtin_amdgcn_wmma_f32_16x16x4_f32(false, a[k], false, b0,
                                                 (short)0, acc0, false, false);
    acc1 = __builtin_amdgcn_wmma_f32_16x16x4_f32(false, a[k], false, b1,
                                                 (short)0, acc1, false, false);
  }

  // D layout: VGPR j -> row j (lanes 0-15) / row j+8 (lanes 16-31), col lane16.
#pragma unroll
  for (int j = 0; j < 8; ++j) {
    const int r = row0 + j + 8 * halfSel;
    if (r < N) {
      const float nv = norm[r];
      hw[(size_t)r * D + lane16]      = acc0[j] * nv;
      hw[(size_t)r * D + 16 + lane16] = acc1[j] * nv;
    }
  }
}

// ---------------------------------------------------------------------------
// Kernel 0: zero the accumulator (must run every call; harness doesn't reset ws)
// ---------------------------------------------------------------------------
__global__ __launch_bounds__(256) void gde_zero(float4* __restrict__ p, int n4)
{
  const int i = blockIdx.x * blockDim.x + threadIdx.x;
  if (i < n4) p[i] = make_float4(0.f, 0.f, 0.f, 0.f);
}

// ---------------------------------------------------------------------------
// Kernel 2: edge scatter-add.  lane == channel (coalesced 128B per wave);
// edge indices forced into SGPRs so src/dst reads become scalar loads.
// hw (12.8 MB) + agg (12.8 MB) both live in the 192 MB L2, so the 51.2M
// fp32 atomics resolve at L2 atomic throughput. 4 edges per wave for ILP.
// ---------------------------------------------------------------------------
__global__ __launch_bounds__(256) void gde_edges(
    const float* __restrict__ hw,   // [N, 32]
    const int* __restrict__ src, const int* __restrict__ dst,
    float* __restrict__ agg,        // [N, 32]
    int E)
{
  const int lane = threadIdx.x & 31;
  int wave = (int)((blockIdx.x * (unsigned)blockDim.x + threadIdx.x) >> 5);
  wave = __builtin_amdgcn_readfirstlane(wave);
  const int e0 = wave * 4;
#pragma unroll
  for (int i = 0; i < 4; ++i) {
    const int e = e0 + i;
    if (e < E) {  // wave-uniform branch
      const int s = __builtin_amdgcn_readfirstlane(src[e]);
      const int d = __builtin_amdgcn_readfirstlane(dst[e]);
      const float v = hw[(size_t)s * D + lane];
      atomicAdd(&agg[(size_t)d * D + lane], v);  // global_atomic_add_f32, no-return
    }
  }
}

// ---------------------------------------------------------------------------
// Kernel 3: out = relu(agg * norm * t + bias)
// ---------------------------------------------------------------------------
__global__ __launch_bounds__(256) void gde_finalize(
    const float* __restrict__ agg, const float* __restrict__ norm,
    const float* __restrict__ bias, const float* __restrict__ tptr,
    float* __restrict__ out, int N)
{
  const int i = blockIdx.x * blockDim.x + threadIdx.x;
  if (i >= N * D) return;
  const int n = i >> 5;
  const int c = i & 31;
  const float t = tptr[0];
  const float v = agg[i] * norm[n] * t + bias[c];
  out[i] = fmaxf(v, 0.0f);
}

// ---------------------------------------------------------------------------
extern "C" void kernel_launch(void* const* d_in, const int* in_sizes, int n_in,
                              void* d_out, int out_size, void* d_ws, size_t ws_size,
                              hipStream_t stream)
{
  const float* t_p    = (const float*)d_in[0];
  const float* h      = (const float*)d_in[1];
  const float* norm   = (const float*)d_in[2];
  const float* weight = (const float*)d_in[3];
  const float* bias   = (const float*)d_in[4];
  const int*   src    = (const int*)d_in[5];
  const int*   dst    = (const int*)d_in[6];
  const int N = in_sizes[2];   // norm is [N,1] -> N elements
  const int E = in_sizes[5];
  float* out = (float*)d_out;

  // Workspace layout: hw [N*32] | agg [N*32]   (25.6 MB total)
  float* hw  = (float*)d_ws;
  float* agg = hw + (size_t)N * D;

  // 0) zero accumulator
  {
    const int n4 = (N * D) / 4;   // N*32 always divisible by 4
    gde_zero<<<(n4 + 255) / 256, 256, 0, stream>>>((float4*)agg, n4);
  }
  // 1) WMMA projection + pre-normalization
  {
    const int numTiles = (N + 15) / 16;
    const int blocks   = (numTiles + 7) / 8;   // 8 waves per 256-thread block
    gde_gemm_wmma<<<blocks, 256, 0, stream>>>(h, weight, norm, hw, N);
  }
  // 2) edge scatter-add (4 edges per wave, 8 waves per block)
  {
    const int waves  = (E + 3) / 4;
    const int blocks = (waves + 7) / 8;
    gde_edges<<<blocks, 256, 0, stream>>>(hw, src, dst, agg, E);
  }
  // 3) epilogue
  {
    const int total = N * D;
    gde_finalize<<<(total + 255) / 256, 256, 0, stream>>>(agg, norm, bias, t_p, out, N);
  }
}